// Encoder_32813550141911
// MI455X (gfx1250) — compile-verified
//
#include <hip/hip_runtime.h>
#include <math.h>

typedef float v2f __attribute__((ext_vector_type(2)));
typedef float v8f __attribute__((ext_vector_type(8)));

#define NNODES 50000
#define NEDGES 800000
#define DIN    256
#define DOUT   64
#define NG     3
#define NB     256

// ------------------------------------------------------------------
// WMMA f32 16x16x4, register-blocked over NT column tiles: one wave
// computes a 16 x (16*NT) strip, loading each A fragment once and
// issuing NT WMMAs on it.
// A frag (ISA 7.12.2, 32-bit A 16x4): lanes 0-15 -> M=0..15; VGPR0 holds
// K = k+2*half, VGPR1 holds K = k+2*half+1 (half = lane>=16).
// B frag mirrors the K mapping; C/D: VGPR v, lane l -> M=v+8*(l>>4), N=l&15.
// ------------------------------------------------------------------
template <int NT>
__device__ __forceinline__ void wmma_strip(const float* __restrict__ A,
                                           const float* __restrict__ B,
                                           int lda, int ldb,
                                           int rowBase, int colBase,
                                           int k0, int k1, v8f acc[NT]) {
  const int lane = threadIdx.x & 31;
  const int half = lane >> 4;
  const int l15  = lane & 15;
  const float* arow = A + (size_t)(rowBase + l15) * lda;
  for (int k = k0; k < k1; k += 4) {
    const int ka = k + half * 2;
    v2f a;
    a.x = arow[ka];
    a.y = arow[ka + 1];
    const float* brow0 = B + (size_t)ka * ldb + colBase + l15;
    const float* brow1 = B + (size_t)(ka + 1) * ldb + colBase + l15;
#pragma unroll
    for (int t = 0; t < NT; ++t) {
      v2f b;
      b.x = brow0[t * 16];
      b.y = brow1[t * 16];
      acc[t] = __builtin_amdgcn_wmma_f32_16x16x4_f32(false, a, false, b,
                                                     (short)0, acc[t], false, false);
    }
  }
}

template <int NT>
__device__ __forceinline__ void wmma_store_strip(float* __restrict__ C, int ldc,
                                                 int rowBase, int colBase, v8f acc[NT]) {
  const int lane = threadIdx.x & 31;
  const int half = lane >> 4;
  const int l15  = lane & 15;
#pragma unroll
  for (int t = 0; t < NT; ++t)
#pragma unroll
    for (int v = 0; v < 8; ++v)
      C[(size_t)(rowBase + v + 8 * half) * ldc + colBase + t * 16 + l15] = acc[t][v];
}

// C[M,N] = A[M,K] @ B[K,N]   (M%16==0, N%64==0, K%4==0)
__global__ void wmma_gemm(const float* __restrict__ A, const float* __restrict__ B,
                          float* __restrict__ C, int M, int K, int N) {
  constexpr int NT = 4;
  const int wave   = (int)((blockIdx.x * blockDim.x + threadIdx.x) >> 5);
  const int groupsN = N / (16 * NT);
  const int total   = (M >> 4) * groupsN;
  if (wave >= total) return;
  const int rowBase = (wave / groupsN) << 4;
  const int colBase = (wave % groupsN) * (16 * NT);
  v8f acc[NT] = {};
  wmma_strip<NT>(A, B, K, N, rowBase, colBase, 0, K, acc);
  wmma_store_strip<NT>(C, N, rowBase, colBase, acc);
}

// C = tanh(A @ B + bias[col])  (attention projection)
__global__ void wmma_gemm_tanh(const float* __restrict__ A, const float* __restrict__ B,
                               const float* __restrict__ bias,
                               float* __restrict__ C, int M, int K, int N) {
  constexpr int NT = 4;
  const int wave   = (int)((blockIdx.x * blockDim.x + threadIdx.x) >> 5);
  const int groupsN = N / (16 * NT);
  const int total   = (M >> 4) * groupsN;
  if (wave >= total) return;
  const int rowBase = (wave / groupsN) << 4;
  const int colBase = (wave % groupsN) * (16 * NT);
  v8f acc[NT] = {};
  wmma_strip<NT>(A, B, K, N, rowBase, colBase, 0, K, acc);
  const int l15 = threadIdx.x & 15;
#pragma unroll
  for (int t = 0; t < NT; ++t) {
    const float bcol = bias[colBase + t * 16 + l15];
#pragma unroll
    for (int v = 0; v < 8; ++v) acc[t][v] = tanhf(acc[t][v] + bcol);
  }
  wmma_store_strip<NT>(C, N, rowBase, colBase, acc);
}

// C += A @ B over K-chunk (split-K, C pre-zeroed, atomic accumulate)
__global__ void wmma_gemm_splitk(const float* __restrict__ A, const float* __restrict__ B,
                                 float* __restrict__ C, int M, int K, int N, int kChunk) {
  constexpr int NT = 4;
  const int wave   = (int)((blockIdx.x * blockDim.x + threadIdx.x) >> 5);
  const int groupsN = N / (16 * NT);
  const int total   = (M >> 4) * groupsN;
  if (wave >= total) return;
  const int rowBase = (wave / groupsN) << 4;
  const int colBase = (wave % groupsN) * (16 * NT);
  const int k0 = blockIdx.y * kChunk;
  int k1 = k0 + kChunk; if (k1 > K) k1 = K;
  v8f acc[NT] = {};
  wmma_strip<NT>(A, B, K, N, rowBase, colBase, k0, k1, acc);
  const int lane = threadIdx.x & 31;
  const int half = lane >> 4;
  const int l15  = lane & 15;
#pragma unroll
  for (int t = 0; t < NT; ++t)
#pragma unroll
    for (int v = 0; v < 8; ++v)
      atomicAdd(&C[(size_t)(rowBase + v + 8 * half) * N + colBase + t * 16 + l15], acc[t][v]);
}

// ------------------------------------------------------------------
// Graph scatter / elementwise kernels
// ------------------------------------------------------------------
__global__ void fill_kernel(float* __restrict__ p, long long n, float v) {
  long long i = (long long)blockIdx.x * blockDim.x + threadIdx.x;
  if (i < n) p[i] = v;
}

// deg[g*N + dst[e]] += (ew ? ew[g*E+e] : 1)   over total = G*E flat
__global__ void deg_accum(const int* __restrict__ dst, const float* __restrict__ ew,
                          float* __restrict__ deg, long long total, int E, int N) {
  long long i = (long long)blockIdx.x * blockDim.x + threadIdx.x;
  if (i >= total) return;
  const int g = (int)(i / E);
  const int e = (int)(i % E);
  const float w = ew ? ew[i] : 1.0f;
  atomicAdd(&deg[(size_t)g * N + dst[e]], w);
}

__global__ void rsqrt_kernel(float* __restrict__ p, long long n) {
  long long i = (long long)blockIdx.x * blockDim.x + threadIdx.x;
  if (i >= n) return;
  const float d = p[i];
  p[i] = d > 0.0f ? rsqrtf(d) : 0.0f;
}

// out[n*F+f] = h[n*F+f] * dis[n]^2   (self-loop contribution, norm = dis*1*dis)
__global__ void selfloop_init(const float* __restrict__ h, const float* __restrict__ dis,
                              float* __restrict__ out, int N, int F) {
  long long i = (long long)blockIdx.x * blockDim.x + threadIdx.x;
  if (i >= (long long)N * F) return;
  const int n = (int)(i / F);
  const float d = dis[n];
  out[i] = h[i] * d * d;
}

// wave per edge: out[dst] += h[src] * (dis[src]*ew*dis[dst]); lane strides F
__global__ void agg_edges(const int* __restrict__ srcs, const int* __restrict__ dsts,
                          const float* __restrict__ ew, const float* __restrict__ dis,
                          const float* __restrict__ h, float* __restrict__ out,
                          int E, int F) {
  const int w    = (int)((blockIdx.x * blockDim.x + threadIdx.x) >> 5);
  const int lane = threadIdx.x & 31;
  if (w >= E) return;
  const int s = srcs[w];
  const int d = dsts[w];
  const float wt   = ew ? ew[w] : 1.0f;
  const float norm = dis[s] * wt * dis[d];
  const float* hs = h + (size_t)s * F;
  float* od = out + (size_t)d * F;
  __builtin_prefetch(hs, 0, 3);  // global_prefetch_b8 on gfx1250
  for (int f = lane; f < F; f += 32)
    atomicAdd(&od[f], hs[f] * norm);
}

__global__ void bias_relu(float* __restrict__ out, const float* __restrict__ bias,
                          long long total, int F) {
  long long i = (long long)blockIdx.x * blockDim.x + threadIdx.x;
  if (i >= total) return;
  const float v = out[i] + bias[(int)(i % F)];
  out[i] = v > 0.0f ? v : 0.0f;
}

// wave per node: scores[n*G+g] = sum_h T[n,h]*qf_g[h]   (H = 64)
__global__ void scores_kernel(const float* __restrict__ T, const float* __restrict__ qf_g,
                              float* __restrict__ scores, int N, int G, int g) {
  const int w    = (int)((blockIdx.x * blockDim.x + threadIdx.x) >> 5);
  const int lane = threadIdx.x & 31;
  if (w >= N) return;
  const float* t = T + (size_t)w * 64;
  float s = t[lane] * qf_g[lane] + t[lane + 32] * qf_g[lane + 32];
#pragma unroll
  for (int m = 16; m >= 1; m >>= 1) s += __shfl_xor(s, m, 32);
  if (lane == 0) scores[(size_t)w * G + g] = s;
}

// block per node (64 threads): softmax over G=3, write attn, fuse hidden
__global__ void attn_fused_kernel(const float* __restrict__ scores,
                                  const float* __restrict__ hidden,
                                  float* __restrict__ attn, float* __restrict__ fused,
                                  int N) {
  const int n = blockIdx.x;
  const int f = threadIdx.x;
  const float s0 = scores[(size_t)n * 3 + 0];
  const float s1 = scores[(size_t)n * 3 + 1];
  const float s2 = scores[(size_t)n * 3 + 2];
  const float mx = fmaxf(s0, fmaxf(s1, s2));
  const float e0 = expf(s0 - mx), e1 = expf(s1 - mx), e2 = expf(s2 - mx);
  const float inv = 1.0f / (e0 + e1 + e2);
  const float a0 = e0 * inv, a1 = e1 * inv, a2 = e2 * inv;
  const size_t NF = (size_t)N * 64;
  const size_t base = (size_t)n * 64 + f;
  fused[base] = a0 * hidden[base] + a1 * hidden[NF + base] + a2 * hidden[2 * NF + base];
  if (f == 0) attn[(size_t)n * 3 + 0] = a0;
  if (f == 1) attn[(size_t)n * 3 + 1] = a1;
  if (f == 2) attn[(size_t)n * 3 + 2] = a2;
}

// block per row b: rsum[b] = sum_n gn[b,n]
__global__ void rowsum_kernel(const float* __restrict__ gn, float* __restrict__ rs, int N) {
  __shared__ float sm[256];
  const int b = blockIdx.x;
  float s = 0.0f;
  for (int i = threadIdx.x; i < N; i += 256) s += gn[(size_t)b * N + i];
  sm[threadIdx.x] = s;
  __syncthreads();
  for (int st = 128; st > 0; st >>= 1) {
    if (threadIdx.x < st) sm[threadIdx.x] += sm[threadIdx.x + st];
    __syncthreads();
  }
  if (threadIdx.x == 0) rs[b] = sm[0];
}

// block per row b (64 threads): g = (vsum/rsum) / max(||.||2, 1e-12)
__global__ void g_norm_kernel(const float* __restrict__ vsum, const float* __restrict__ rs,
                              float* __restrict__ gout) {
  __shared__ float sm[64];
  const int b = blockIdx.x;
  const int f = threadIdx.x;
  const float v = vsum[(size_t)b * 64 + f] / rs[b];
  sm[f] = v * v;
  __syncthreads();
  for (int st = 32; st > 0; st >>= 1) {
    if (f < st) sm[f] += sm[f + st];
    __syncthreads();
  }
  const float nrm = fmaxf(sqrtf(sm[0]), 1e-12f);
  gout[(size_t)b * 64 + f] = v / nrm;
}

// ------------------------------------------------------------------
static inline int cdiv(long long a, long long b) { return (int)((a + b - 1) / b); }

extern "C" void kernel_launch(void* const* d_in, const int* in_sizes, int n_in,
                              void* d_out, int out_size, void* d_ws, size_t ws_size,
                              hipStream_t stream) {
  const float* x  = (const float*)d_in[0];
  const int*   ei = (const int*)d_in[1];   // int32 (jax x64 disabled)
  const float* gn = (const float*)d_in[2];
  const float* ew = (const float*)d_in[3];
  const float* W1 = (const float*)d_in[4];
  const float* b1 = (const float*)d_in[5];
  const float* W2 = (const float*)d_in[6];
  const float* b2 = (const float*)d_in[7];
  const float* Wf = (const float*)d_in[8];
  const float* bf = (const float*)d_in[9];
  const float* qf = (const float*)d_in[10];
  (void)in_sizes; (void)n_in; (void)out_size; (void)ws_size;

  const int* srcs = ei;
  const int* dsts = ei + NEDGES;

  // output layout: fused | emb | g | hidden | attn
  float* out    = (float*)d_out;
  float* fused  = out;
  float* emb    = fused + (size_t)NNODES * DOUT;
  float* gout   = emb   + (size_t)NNODES * DIN;
  float* hidden = gout  + (size_t)NB * DOUT;
  float* attn   = hidden + (size_t)NG * NNODES * DOUT;

  // workspace layout (floats): bufA(h1 then h2) | T | dis | dis2 | scores | vsum | rsum
  float* ws     = (float*)d_ws;
  float* bufA   = ws;                                   // NNODES*DIN (h1 uses first NNODES*DOUT)
  float* T      = bufA + (size_t)NNODES * DIN;          // NNODES*DOUT
  float* dis    = T + (size_t)NNODES * DOUT;            // NG*NNODES
  float* dis2   = dis + (size_t)NG * NNODES;            // NNODES
  float* scores = dis2 + NNODES;                        // NNODES*NG
  float* vsum   = scores + (size_t)NNODES * NG;         // NB*DOUT
  float* rsum   = vsum + (size_t)NB * DOUT;             // NB

  const int BLK = 256, WPB = BLK / 32;
  const size_t NF = (size_t)NNODES * DOUT;

  // 1) h1 = x @ W1  [50000 x 256 x 64]; one wave per 16x64 strip
  {
    const int tiles = (NNODES / 16) * (DOUT / 64);
    wmma_gemm<<<cdiv(tiles, WPB), BLK, 0, stream>>>(x, W1, bufA, NNODES, DIN, DOUT);
  }

  // 2) degrees + rsqrt for the 3 weighted graphs (self-loop weight 1)
  fill_kernel<<<cdiv((long long)NG * NNODES, BLK), BLK, 0, stream>>>(dis, (long long)NG * NNODES, 1.0f);
  deg_accum<<<cdiv((long long)NG * NEDGES, BLK), BLK, 0, stream>>>(dsts, ew, dis,
                                                                   (long long)NG * NEDGES, NEDGES, NNODES);
  rsqrt_kernel<<<cdiv((long long)NG * NNODES, BLK), BLK, 0, stream>>>(dis, (long long)NG * NNODES);

  // 3) hidden[g] = selfloop + edge scatter of h1, then +b1, relu
  for (int g = 0; g < NG; ++g) {
    selfloop_init<<<cdiv((long long)NNODES * DOUT, BLK), BLK, 0, stream>>>(
        bufA, dis + (size_t)g * NNODES, hidden + (size_t)g * NF, NNODES, DOUT);
    agg_edges<<<cdiv((long long)NEDGES * 32, BLK), BLK, 0, stream>>>(
        srcs, dsts, ew + (size_t)g * NEDGES, dis + (size_t)g * NNODES,
        bufA, hidden + (size_t)g * NF, NEDGES, DOUT);
  }
  bias_relu<<<cdiv((long long)NG * NNODES * DOUT, BLK), BLK, 0, stream>>>(
      hidden, b1, (long long)NG * NNODES * DOUT, DOUT);

  // 4) attention: T = tanh(hidden_g @ Wf_g + bf_g); scores = T . qf_g; softmax; fuse
  for (int g = 0; g < NG; ++g) {
    const int tiles = (NNODES / 16) * (DOUT / 64);
    wmma_gemm_tanh<<<cdiv(tiles, WPB), BLK, 0, stream>>>(
        hidden + (size_t)g * NF, Wf + (size_t)g * DOUT * DOUT, bf + (size_t)g * DOUT,
        T, NNODES, DOUT, DOUT);
    scores_kernel<<<cdiv((long long)NNODES * 32, BLK), BLK, 0, stream>>>(
        T, qf + (size_t)g * DOUT, scores, NNODES, NG, g);
  }
  attn_fused_kernel<<<NNODES, 64, 0, stream>>>(scores, hidden, attn, fused, NNODES);

  // 5) h2 = fused @ W2  [50000 x 64 x 256]; one wave per 16x64 strip
  {
    const int tiles = (NNODES / 16) * (DIN / 64);
    wmma_gemm<<<cdiv(tiles, WPB), BLK, 0, stream>>>(fused, W2, bufA, NNODES, DOUT, DIN);
  }

  // 6) unweighted degree (+self loop) + rsqrt
  fill_kernel<<<cdiv((long long)NNODES, BLK), BLK, 0, stream>>>(dis2, NNODES, 1.0f);
  deg_accum<<<cdiv((long long)NEDGES, BLK), BLK, 0, stream>>>(dsts, (const float*)nullptr, dis2,
                                                              (long long)NEDGES, NEDGES, NNODES);
  rsqrt_kernel<<<cdiv((long long)NNODES, BLK), BLK, 0, stream>>>(dis2, NNODES);

  // 7) emb = relu(scatter(h2) + b2)   [50000 x 256]
  selfloop_init<<<cdiv((long long)NNODES * DIN, BLK), BLK, 0, stream>>>(bufA, dis2, emb, NNODES, DIN);
  agg_edges<<<cdiv((long long)NEDGES * 32, BLK), BLK, 0, stream>>>(
      srcs, dsts, (const float*)nullptr, dis2, bufA, emb, NEDGES, DIN);
  bias_relu<<<cdiv((long long)NNODES * DIN, BLK), BLK, 0, stream>>>(
      emb, b2, (long long)NNODES * DIN, DIN);

  // 8) vsum = graph_neigh @ fused (split-K WMMA), then masked-avg + L2 normalize
  fill_kernel<<<cdiv((long long)NB * DOUT, BLK), BLK, 0, stream>>>(vsum, (long long)NB * DOUT, 0.0f);
  {
    const int tiles  = (NB / 16) * (DOUT / 64);        // 16 waves per chunk
    const int chunks = 100, kChunk = NNODES / chunks;  // 500, %4==0
    dim3 grid(cdiv(tiles, WPB), chunks);
    wmma_gemm_splitk<<<grid, BLK, 0, stream>>>(gn, fused, vsum, NB, NNODES, DOUT, kChunk);
  }
  rowsum_kernel<<<NB, 256, 0, stream>>>(gn, rsum, NNODES);
  g_norm_kernel<<<NB, 64, 0, stream>>>(vsum, rsum, gout);
}